// PtModule_76166950028098
// MI455X (gfx1250) — compile-verified
//
#include <hip/hip_runtime.h>

// Elementwise y = 2x + 5 - 3/(1+x) over 8192^2 f32 (64Mi elements).
// Memory-bound: 537 MB mandatory traffic -> ~23 us floor at 23.3 TB/s.
// Shape: one-shot grid (no grid-stride), 32-bit offsets so loads use the
// saddr+voffset form with immediate IOFFSETs, 4x b128 NT loads/stores per
// thread for MLP, v_rcp_f32 fast reciprocal.

typedef float v4f __attribute__((ext_vector_type(4)));

__device__ __forceinline__ v4f ew_op(v4f x) {
    v4f r;
#pragma unroll
    for (int k = 0; k < 4; ++k) {
        float xv  = x[k];
        float rcp = __builtin_amdgcn_rcpf(1.0f + xv);        // v_rcp_f32
        r[k] = __builtin_fmaf(xv, 2.0f, 5.0f) - 3.0f * rcp;  // 2 FMAs
    }
    return r;
}

// Each 256-thread block (8 wave32) covers 1024 float4 tiles = 16 KiB in,
// 16 KiB out. Intra-thread tile stride is the constant 256 (=4096 bytes),
// so loads 1..3 fold into the 24-bit instruction offset of one vaddr.
__global__ __launch_bounds__(256) void ew_main(const v4f* __restrict__ in4,
                                               v4f* __restrict__ out4,
                                               unsigned n4) {
    const unsigned base = blockIdx.x * 1024u + threadIdx.x;
    const unsigned i0 = base;
    const unsigned i1 = base + 256u;
    const unsigned i2 = base + 512u;
    const unsigned i3 = base + 768u;

    if (i3 < n4) {  // fast path: all blocks except possibly the last
        v4f a = __builtin_nontemporal_load(in4 + i0);
        v4f b = __builtin_nontemporal_load(in4 + i1);
        v4f c = __builtin_nontemporal_load(in4 + i2);
        v4f d = __builtin_nontemporal_load(in4 + i3);
        __builtin_nontemporal_store(ew_op(a), out4 + i0);
        __builtin_nontemporal_store(ew_op(b), out4 + i1);
        __builtin_nontemporal_store(ew_op(c), out4 + i2);
        __builtin_nontemporal_store(ew_op(d), out4 + i3);
    } else {        // boundary block only
        if (i0 < n4) {
            v4f a = __builtin_nontemporal_load(in4 + i0);
            __builtin_nontemporal_store(ew_op(a), out4 + i0);
        }
        if (i1 < n4) {
            v4f b = __builtin_nontemporal_load(in4 + i1);
            __builtin_nontemporal_store(ew_op(b), out4 + i1);
        }
        if (i2 < n4) {
            v4f c = __builtin_nontemporal_load(in4 + i2);
            __builtin_nontemporal_store(ew_op(c), out4 + i2);
        }
    }
}

// Scalar tail for n % 4 != 0 (never hit for 8192^2; kept for generality).
__global__ __launch_bounds__(32) void ew_tail(const float* __restrict__ in,
                                              float* __restrict__ out,
                                              unsigned base, unsigned n) {
    unsigned i = base + threadIdx.x;
    if (i < n) {
        float x = in[i];
        out[i] = __builtin_fmaf(x, 2.0f, 5.0f) -
                 3.0f * __builtin_amdgcn_rcpf(1.0f + x);
    }
}

extern "C" void kernel_launch(void* const* d_in, const int* in_sizes, int n_in,
                              void* d_out, int out_size, void* d_ws, size_t ws_size,
                              hipStream_t stream) {
    const float* in  = (const float*)d_in[0];
    float*       out = (float*)d_out;

    const unsigned n   = (unsigned)out_size;  // 67,108,864
    const unsigned n4  = n >> 2;              // 16,777,216 float4 tiles
    const unsigned rem = n & 3u;

    const unsigned blocks = (n4 + 1023u) / 1024u;  // 16384 for 8192^2
    if (blocks > 0) {
        ew_main<<<blocks, 256, 0, stream>>>((const v4f*)in, (v4f*)out, n4);
    }
    if (rem) {
        ew_tail<<<1, 32, 0, stream>>>(in, out, n4 << 2, n);
    }
}